// TemporalShift_50242527428854
// MI455X (gfx1250) — compile-verified
//
#include <hip/hip_runtime.h>

// Temporal Shift Module (TSM) for x: (nt=64, c=256, h=56, w=56) f32.
// Viewed as (n=8, t=8, c, h, w):
//   ch in [0,32):   out[t] = x[t+1]  (zero at t==7)
//   ch in [32,64):  out[t] = x[t-1]  (zero at t==0)
//   ch in [64,256): out[t] = x[t]
//
// Pure permutation, 0 FLOPs, ~411 MB traffic -> HBM-bound (~17.6 us at
// 23.3 TB/s). One block per (nt,c) plane: all shift/index logic is scalar
// (SALU, uniform per block); threads stream 784 float4s per plane with
// b128 non-temporal loads/stores (working set 2x205 MB > 192 MB L2).

typedef float v4f __attribute__((ext_vector_type(4)));

#define TSM_HW4   784   // (56*56)/4 float4 elements per (nt,c) plane
#define TSM_C     256
#define TSM_T     8
#define TSM_FOLD  32

__global__ __launch_bounds__(256) void tsm_shift_kernel(
    const v4f* __restrict__ x, v4f* __restrict__ out)
{
    const int plane = blockIdx.x;              // nt*C + c : 0..16383 (scalar)
    const int tid   = threadIdx.x;

    const int ch = plane & (TSM_C - 1);        // C = 256 (pow2)
    const int t  = (plane >> 8) & (TSM_T - 1); // T = 8   (pow2)

    // Plane-uniform shift decision (pure SALU + scalar branch).
    int  src_plane = plane;
    bool zero      = false;
    if (ch < TSM_FOLD) {                       // take from t+1
        if (t == TSM_T - 1) zero = true;
        else                src_plane = plane + TSM_C;
    } else if (ch < 2 * TSM_FOLD) {            // take from t-1
        if (t == 0) zero = true;
        else        src_plane = plane - TSM_C;
    }

    v4f* dst = out + (size_t)plane * TSM_HW4;

    if (zero) {
        const v4f z = (v4f){0.f, 0.f, 0.f, 0.f};
        __builtin_nontemporal_store(z, dst + tid);
        __builtin_nontemporal_store(z, dst + 256 + tid);
        __builtin_nontemporal_store(z, dst + 512 + tid);
        if (tid < TSM_HW4 - 768)               // tail: 16 float4s
            __builtin_nontemporal_store(z, dst + 768 + tid);
    } else {
        const v4f* src = x + (size_t)src_plane * TSM_HW4;
        // Issue all loads first: 3 outstanding b128 loads per lane for
        // latency hiding, then drain with stores.
        v4f v0 = __builtin_nontemporal_load(src + tid);
        v4f v1 = __builtin_nontemporal_load(src + 256 + tid);
        v4f v2 = __builtin_nontemporal_load(src + 512 + tid);
        __builtin_nontemporal_store(v0, dst + tid);
        __builtin_nontemporal_store(v1, dst + 256 + tid);
        __builtin_nontemporal_store(v2, dst + 512 + tid);
        if (tid < TSM_HW4 - 768) {             // tail: 16 float4s
            v4f v3 = __builtin_nontemporal_load(src + 768 + tid);
            __builtin_nontemporal_store(v3, dst + 768 + tid);
        }
    }
}

extern "C" void kernel_launch(void* const* d_in, const int* in_sizes, int n_in,
                              void* d_out, int out_size, void* d_ws, size_t ws_size,
                              hipStream_t stream)
{
    const v4f* x   = (const v4f*)d_in[0];
    v4f*       out = (v4f*)d_out;

    // planes = nt * c = total / (h*w) = 51,380,224 / 3136 = 16384
    const int planes = in_sizes[0] / (TSM_HW4 * 4);

    tsm_shift_kernel<<<planes, 256, 0, stream>>>(x, out);
}